// GNNPolicy_66580583022718
// MI455X (gfx1250) — compile-verified
//
#include <hip/hip_runtime.h>

// ===========================================================================
// GNNPolicy forward for MI455X (gfx1250, wave32, WMMA).
//
// Pipeline (all fp32, matching the fp32 reference):
//   cons = MLP(LN(cons_x))          var = MLP(LN(var_x))
//   v2c bipartite conv  ->  new cons
//   c2v bipartite conv  ->  new var
//   out = relu(var @ Wout1 + b) @ Wout2
//
// Roofline: ~16.4 GFLOP of per-edge 16x64 @ 64x64 GEMM (fp32 WMMA 16x16x4)
// vs ~1.5 GB of L2-resident gather/scatter traffic (node tables are 12.8 /
// 25.6 MB << 192 MB L2). Memory-bound in L2 -> fuse everything per edge,
// never materialize the E x 64 message tensor, keep fp32 (matrix ceiling
// irrelevant here, precision matches the fp32 reference).
//
// v2: all GEMM weights are pre-transposed (64 x K, column-pair contiguous)
// so every WMMA B-fragment is ONE global_load_b64 / ds_load_b64 instead of
// two b32 loads; the edge kernel stages Wf^T + LN affine + bias + the folded
// constant edge vector in LDS (17 KB) so 8 waves/block read weights via DS
// instead of re-pulling 16 KB/wave from L1.
//
// NOTE on _ln(edge_features): LayerNorm over a single feature is exactly the
// bias (x - mean(x) == 0 bit-exactly), so ef @ We == edge_ln_b[0] * We[:],
// folded into the LDS-staged constant vector `lef`.
//
// Assumed d_in flattening (setup_inputs dict insertion order at top level,
// nested params as JAX tree leaves = sorted keys, ASCII order W* < b* < ...):
//   0 constraint_features  1 edge_features  2 variable_features
//   3 Wout1  4 Wout2  5 bout1
//   6..19  c2v: We Wf Wl Wo1 Wo2 Wr bf bl bo1 bo2 fln_b fln_g pln_b pln_g
//   20..25 cons embed: W1 W2 b1 b2 ln_b ln_g
//   26 edge_ln_b  27 edge_ln_g
//   28..41 v2c: (same order as c2v)
//   42..47 var embed: W1 W2 b1 b2 ln_b ln_g
//   48 edge_indices (int64, [2,E])   49 n_var
// ===========================================================================

typedef __attribute__((ext_vector_type(2))) float v2f;
typedef __attribute__((ext_vector_type(4))) float v4f;
typedef __attribute__((ext_vector_type(8))) float v8f;

#define LN_EPS 1e-5f

static __device__ __forceinline__ v8f wmma4(v2f a, v2f b, v8f c) {
  // V_WMMA_F32_16X16X4_F32 : D(16x16 f32) = A(16x4 f32) x B(4x16 f32) + C
  return __builtin_amdgcn_wmma_f32_16x16x4_f32(
      /*neg_a=*/false, a, /*neg_b=*/false, b,
      /*c_mod=*/(short)0, c, /*reuse_a=*/false, /*reuse_b=*/false);
}

// ---------------------------------------------------------------------------
__global__ void fill0(float* __restrict__ p, int n) {
  int i = blockIdx.x * blockDim.x + threadIdx.x;
  if (i < n) p[i] = 0.f;
}

// ---------------------------------------------------------------------------
// Weight transpose: in (K x 64, row-major) -> out (64 x K): out[n*K+k]=in[k*64+n]
// Makes the WMMA B-fragment pair {W[k][n], W[k+1][n]} contiguous (one b64 load).
__global__ void transposeW(const float* __restrict__ in, float* __restrict__ out, int K) {
  int i = blockIdx.x * blockDim.x + threadIdx.x;
  if (i < K * 64) {
    int k = i >> 6;      // i / 64
    int n = i & 63;      // i % 64
    out[n * K + k] = in[i];
  }
}

// ---------------------------------------------------------------------------
// Embed stage 1: out[N,64] = relu( LN(x[N,NF]) @ W1(NF,64) + b1 )
// One block (64 threads) per node; NF is tiny (5 or 19).
template <int NF>
__global__ void embed_stage1(const float* __restrict__ x,
                             const float* __restrict__ g,
                             const float* __restrict__ b,
                             const float* __restrict__ W1,
                             const float* __restrict__ b1,
                             float* __restrict__ out, int N) {
  __shared__ float xs[NF];
  int node = blockIdx.x;
  if (node >= N) return;
  int j = threadIdx.x;  // 0..63
  if (j < NF) xs[j] = x[(size_t)node * NF + j];
  __syncthreads();
  float m = 0.f;
#pragma unroll
  for (int k = 0; k < NF; ++k) m += xs[k];
  m *= (1.f / NF);
  float v = 0.f;
#pragma unroll
  for (int k = 0; k < NF; ++k) { float d = xs[k] - m; v += d * d; }
  v *= (1.f / NF);
  float rstd = rsqrtf(v + LN_EPS);
  float acc = b1[j];
#pragma unroll
  for (int k = 0; k < NF; ++k) {
    float ln = (xs[k] - m) * rstd * g[k] + b[k];
    acc += ln * W1[k * 64 + j];
  }
  out[(size_t)node * 64 + j] = fmaxf(acc, 0.f);
}

// ---------------------------------------------------------------------------
// Rowwise LayerNorm over 64 features: out[N,64] = LN(in)*g + b
__global__ void ln64(const float* __restrict__ in, const float* __restrict__ g,
                     const float* __restrict__ b, float* __restrict__ out, int N) {
  __shared__ float xs[64];
  int row = blockIdx.x;
  if (row >= N) return;
  int t = threadIdx.x;
  float v = in[(size_t)row * 64 + t];
  xs[t] = v;
  __syncthreads();
  float m = 0.f;
#pragma unroll
  for (int k = 0; k < 64; ++k) m += xs[k];
  m *= (1.f / 64.f);
  float var = 0.f;
#pragma unroll
  for (int k = 0; k < 64; ++k) { float d = xs[k] - m; var += d * d; }
  var *= (1.f / 64.f);
  out[(size_t)row * 64 + t] = (v - m) * rsqrtf(var + LN_EPS) * g[t] + b[t];
}

// ---------------------------------------------------------------------------
// WMMA row-tile GEMM: out[N,64] = act( [A0 | A1](N, K0+K1) @ W(K0+K1,64) + b )
// WT is the TRANSPOSED weight (64 x K): WT[n*K + k] = W[k][n], so each
// B-fragment {W[k][n], W[k+1][n]} is a single contiguous b64 load.
// One wave per 16-row tile; wave-uniform guards keep EXEC all-ones at WMMA.
template <int K0, int K1, bool RELU, bool BIAS>
__global__ void wmma_gemm(const float* __restrict__ A0,
                          const float* __restrict__ A1,
                          const float* __restrict__ WT,
                          const float* __restrict__ bias,
                          float* __restrict__ out, int N) {
  int tile = blockIdx.x * (blockDim.x >> 5) + (threadIdx.x >> 5);
  int numTiles = (N + 15) >> 4;
  if (tile >= numTiles) return;  // wave-uniform
  int lane = threadIdx.x & 31;
  int hi = lane >> 4;     // K-half selector
  int n  = lane & 15;     // also the A-matrix row m for this lane
  int rowBase = tile << 4;
  int rowc = min(rowBase + n, N - 1);
  constexpr int K = K0 + K1;

  const float* wc0 = WT + (size_t)(0 * 16 + n) * K;
  const float* wc1 = WT + (size_t)(1 * 16 + n) * K;
  const float* wc2 = WT + (size_t)(2 * 16 + n) * K;
  const float* wc3 = WT + (size_t)(3 * 16 + n) * K;

  v8f acc[4] = {};
#pragma unroll
  for (int t = 0; t < K / 4; ++t) {
    int k = 4 * t + 2 * hi;
    v2f a;
    if (K1 == 0 || k < K0) {
      a = *(const v2f*)(A0 + (size_t)rowc * K0 + k);
    } else {
      a = *(const v2f*)(A1 + (size_t)rowc * K1 + (k - K0));
    }
    v2f b0 = *(const v2f*)(wc0 + k);
    v2f b1 = *(const v2f*)(wc1 + k);
    v2f b2 = *(const v2f*)(wc2 + k);
    v2f b3 = *(const v2f*)(wc3 + k);
    acc[0] = wmma4(a, b0, acc[0]);
    acc[1] = wmma4(a, b1, acc[1]);
    acc[2] = wmma4(a, b2, acc[2]);
    acc[3] = wmma4(a, b3, acc[3]);
  }

#pragma unroll
  for (int c = 0; c < 4; ++c) {
    float bv = BIAS ? bias[c * 16 + n] : 0.f;
#pragma unroll
    for (int r = 0; r < 8; ++r) {
      int row = rowBase + r + 8 * hi;
      if (row < N) {
        float val = acc[c][r] + bv;
        if (RELU) val = fmaxf(val, 0.f);
        out[(size_t)row * 64 + c * 16 + n] = val;
      }
    }
  }
}

// ---------------------------------------------------------------------------
// Fused edge kernel: one wave per 16 edges, 8 waves per block.
//   stage:  Wf^T (16 KB) + fln_g/fln_b/bf + lef = edge_ln_b[0]*We  -> LDS
//   m = li[tgt] + rj[src] + lef                  (gather, L2 resident)
//   a = relu(LN(m) * fln_g + fln_b)              (shfl_xor(16) reduction)
//   d = a(16x64) @ Wf(64x64) + bf                (64x WMMA f32 16x16x4,
//                                                 B-frags via ds_load_b64)
//   agg[tgt] += d                                (global f32 atomics)
__global__ void edge_conv(const float* __restrict__ li, const float* __restrict__ rj,
                          const long long* __restrict__ tgt,
                          const long long* __restrict__ src,
                          const float* __restrict__ edge_ln_b,
                          const float* __restrict__ We,
                          const float* __restrict__ fln_g,
                          const float* __restrict__ fln_b,
                          const float* __restrict__ WfT,  // 64 x 64 transposed
                          const float* __restrict__ bf,
                          float* __restrict__ agg, int E) {
  __shared__ float lwf[64 * 64];
  __shared__ float lg[64];
  __shared__ float lb[64];
  __shared__ float lbf[64];
  __shared__ float lef[64];

  // ---- stage weights into LDS (before any wave can exit) ----
  int tid = threadIdx.x;
  for (int i = tid; i < 1024; i += 256) {
    ((v4f*)lwf)[i] = ((const v4f*)WfT)[i];
  }
  if (tid < 64) {
    lg[tid]  = fln_g[tid];
    lb[tid]  = fln_b[tid];
    lbf[tid] = bf[tid];
    lef[tid] = edge_ln_b[0] * We[tid];
  }
  __syncthreads();

  int tile = blockIdx.x * (blockDim.x >> 5) + (tid >> 5);
  int numTiles = (E + 15) >> 4;
  if (tile >= numTiles) return;  // wave-uniform
  int lane = tid & 31;
  int hi = lane >> 4;
  int n  = lane & 15;            // edge-in-tile for gather, N-column for GEMM
  int e  = tile * 16 + n;
  int ec = min(e, E - 1);
  size_t ti = (size_t)tgt[ec];
  size_t si = (size_t)src[ec];

  // Lane holds 32 of the edge's 64 features: f = 4t + 2*hi + {0,1}
  float mv[32];
  float s1 = 0.f, s2 = 0.f;
#pragma unroll
  for (int t = 0; t < 16; ++t) {
    int f = 4 * t + 2 * hi;
    v2f lv = *(const v2f*)(li + ti * 64 + f);
    v2f rv = *(const v2f*)(rj + si * 64 + f);
    v2f ev = *(const v2f*)(lef + f);
    float m0 = lv.x + rv.x + ev.x;
    float m1 = lv.y + rv.y + ev.y;
    mv[2 * t] = m0;
    mv[2 * t + 1] = m1;
    s1 += m0 + m1;
    s2 += m0 * m0 + m1 * m1;
  }
  // combine the two half-rows (lanes e and e+16) of this edge
  s1 += __shfl_xor(s1, 16, 32);
  s2 += __shfl_xor(s2, 16, 32);
  float mean = s1 * (1.f / 64.f);
  float var = s2 * (1.f / 64.f) - mean * mean;
  float rstd = rsqrtf(fmaxf(var, 0.f) + LN_EPS);
#pragma unroll
  for (int t = 0; t < 16; ++t) {
    int f = 4 * t + 2 * hi;
    v2f gv = *(const v2f*)(lg + f);
    v2f bv = *(const v2f*)(lb + f);
    mv[2 * t]     = fmaxf((mv[2 * t]     - mean) * rstd * gv.x + bv.x, 0.f);
    mv[2 * t + 1] = fmaxf((mv[2 * t + 1] - mean) * rstd * gv.y + bv.y, 0.f);
  }

  // 16x64 @ 64x64 via WMMA f32 16x16x4 (A frags already in layout,
  // B frags contiguous in LDS: lwf[n'*64 + k] = Wf[k][n'])
  const float* wc0 = lwf + (0 * 16 + n) * 64;
  const float* wc1 = lwf + (1 * 16 + n) * 64;
  const float* wc2 = lwf + (2 * 16 + n) * 64;
  const float* wc3 = lwf + (3 * 16 + n) * 64;
  v8f acc[4] = {};
#pragma unroll
  for (int t = 0; t < 16; ++t) {
    int k = 4 * t + 2 * hi;
    v2f a = {mv[2 * t], mv[2 * t + 1]};
    v2f b0 = *(const v2f*)(wc0 + k);
    v2f b1 = *(const v2f*)(wc1 + k);
    v2f b2 = *(const v2f*)(wc2 + k);
    v2f b3 = *(const v2f*)(wc3 + k);
    acc[0] = wmma4(a, b0, acc[0]);
    acc[1] = wmma4(a, b1, acc[1]);
    acc[2] = wmma4(a, b2, acc[2]);
    acc[3] = wmma4(a, b3, acc[3]);
  }

  float bfv[4];
#pragma unroll
  for (int c = 0; c < 4; ++c) bfv[c] = lbf[c * 16 + n];

  // D layout: lane holds D[r + 8*hi][n]; scatter-add into agg[tgt_row]
#pragma unroll
  for (int r = 0; r < 8; ++r) {
    int er = tile * 16 + r + 8 * hi;
    bool ok = er < E;
    size_t tr = (size_t)tgt[min(er, E - 1)];
#pragma unroll
    for (int c = 0; c < 4; ++c) {
      if (ok) atomicAdd(agg + tr * 64 + c * 16 + n, acc[c][r] + bfv[c]);
    }
  }
}

// ---------------------------------------------------------------------------
// out[n] = h[n,:] . Wout2[:,0]
__global__ void out_dot(const float* __restrict__ h, const float* __restrict__ w,
                        float* __restrict__ out, int N) {
  int i = blockIdx.x * blockDim.x + threadIdx.x;
  if (i >= N) return;
  const float* r = h + (size_t)i * 64;
  float acc = 0.f;
#pragma unroll
  for (int k = 0; k < 64; ++k) acc += r[k] * w[k];
  out[i] = acc;
}

// ===========================================================================
struct ConvP {
  const float *We, *Wf, *Wl, *Wo1, *Wo2, *Wr, *bf, *bl, *bo1, *bo2;
  const float *fln_b, *fln_g, *pln_b, *pln_g;
};
static ConvP getConv(void* const* d_in, int base) {
  ConvP p;
  p.We    = (const float*)d_in[base + 0];
  p.Wf    = (const float*)d_in[base + 1];
  p.Wl    = (const float*)d_in[base + 2];
  p.Wo1   = (const float*)d_in[base + 3];
  p.Wo2   = (const float*)d_in[base + 4];
  p.Wr    = (const float*)d_in[base + 5];
  p.bf    = (const float*)d_in[base + 6];
  p.bl    = (const float*)d_in[base + 7];
  p.bo1   = (const float*)d_in[base + 8];
  p.bo2   = (const float*)d_in[base + 9];
  p.fln_b = (const float*)d_in[base + 10];
  p.fln_g = (const float*)d_in[base + 11];
  p.pln_b = (const float*)d_in[base + 12];
  p.pln_g = (const float*)d_in[base + 13];
  return p;
}

extern "C" void kernel_launch(void* const* d_in, const int* in_sizes, int n_in,
                              void* d_out, int out_size, void* d_ws, size_t ws_size,
                              hipStream_t stream) {
  (void)n_in; (void)out_size; (void)ws_size;
  const int NC = in_sizes[0] / 5;
  const int E  = in_sizes[1];
  const int NV = in_sizes[2] / 19;

  const float* cons_x = (const float*)d_in[0];
  const float* var_x  = (const float*)d_in[2];
  const float* Wout1  = (const float*)d_in[3];
  const float* Wout2  = (const float*)d_in[4];
  const float* bout1  = (const float*)d_in[5];
  ConvP c2v = getConv(d_in, 6);
  // cons embed (20..25): W1 W2 b1 b2 ln_b ln_g
  const float* cW1   = (const float*)d_in[20];
  const float* cW2   = (const float*)d_in[21];
  const float* cb1   = (const float*)d_in[22];
  const float* cb2   = (const float*)d_in[23];
  const float* cln_b = (const float*)d_in[24];
  const float* cln_g = (const float*)d_in[25];
  const float* eln_b = (const float*)d_in[26];
  ConvP v2c = getConv(d_in, 28);
  // var embed (42..47)
  const float* vW1   = (const float*)d_in[42];
  const float* vW2   = (const float*)d_in[43];
  const float* vb1   = (const float*)d_in[44];
  const float* vb2   = (const float*)d_in[45];
  const float* vln_b = (const float*)d_in[46];
  const float* vln_g = (const float*)d_in[47];
  const long long* eidx  = (const long long*)d_in[48];
  const long long* c_idx = eidx;       // edge_indices[0]
  const long long* v_idx = eidx + E;   // edge_indices[1]

  // Workspace: 5 big slots (S1 cons-sized, rest var-sized, ~115 MB) plus a
  // small region for transposed weights (61440 floats = 240 KB).
  float* S0 = (float*)d_ws;                 // var_emb -> var2
  float* S1 = S0 + (size_t)NV * 64;         // cons_emb -> cons2
  float* S2 = S1 + (size_t)NC * 64;         // tmp / li_v2c / rj_c2v / t_c2v / h
  float* S3 = S2 + (size_t)NV * 64;         // rj_v2c / li_c2v / ln_agg_c2v
  float* S4 = S3 + (size_t)NV * 64;         // agg / t_v2c
  float* wt = S4 + (size_t)NV * 64;
  float* cW2T     = wt;           wt += 4096;
  float* vW2T     = wt;           wt += 4096;
  float* Wout1T   = wt;           wt += 4096;
  float* v2c_WlT  = wt;           wt += 4096;
  float* v2c_WrT  = wt;           wt += 4096;
  float* v2c_WfT  = wt;           wt += 4096;
  float* v2c_Wo1T = wt;           wt += 8192;   // 64 x 128
  float* v2c_Wo2T = wt;           wt += 4096;
  float* c2v_WlT  = wt;           wt += 4096;
  float* c2v_WrT  = wt;           wt += 4096;
  float* c2v_WfT  = wt;           wt += 4096;
  float* c2v_Wo1T = wt;           wt += 8192;
  float* c2v_Wo2T = wt;           wt += 4096;

  const dim3 blk(256);
  auto gemmGrid = [](int N) { return dim3((unsigned)((((N + 15) / 16) + 7) / 8)); };
  dim3 edgeGrid((unsigned)((((E + 15) / 16) + 7) / 8));
  dim3 tr64((unsigned)((64 * 64 + 255) / 256));
  dim3 tr128((unsigned)((128 * 64 + 255) / 256));

  // ---- Transpose all GEMM weights (contiguous B-fragment pairs) ----
  transposeW<<<tr64, blk, 0, stream>>>(cW2, cW2T, 64);
  transposeW<<<tr64, blk, 0, stream>>>(vW2, vW2T, 64);
  transposeW<<<tr64, blk, 0, stream>>>(Wout1, Wout1T, 64);
  transposeW<<<tr64, blk, 0, stream>>>(v2c.Wl, v2c_WlT, 64);
  transposeW<<<tr64, blk, 0, stream>>>(v2c.Wr, v2c_WrT, 64);
  transposeW<<<tr64, blk, 0, stream>>>(v2c.Wf, v2c_WfT, 64);
  transposeW<<<tr128, blk, 0, stream>>>(v2c.Wo1, v2c_Wo1T, 128);
  transposeW<<<tr64, blk, 0, stream>>>(v2c.Wo2, v2c_Wo2T, 64);
  transposeW<<<tr64, blk, 0, stream>>>(c2v.Wl, c2v_WlT, 64);
  transposeW<<<tr64, blk, 0, stream>>>(c2v.Wr, c2v_WrT, 64);
  transposeW<<<tr64, blk, 0, stream>>>(c2v.Wf, c2v_WfT, 64);
  transposeW<<<tr128, blk, 0, stream>>>(c2v.Wo1, c2v_Wo1T, 128);
  transposeW<<<tr64, blk, 0, stream>>>(c2v.Wo2, c2v_Wo2T, 64);

  // ---- Node embeddings ----
  embed_stage1<5><<<NC, 64, 0, stream>>>(cons_x, cln_g, cln_b, cW1, cb1, S2, NC);
  wmma_gemm<64, 0, true, true><<<gemmGrid(NC), blk, 0, stream>>>(S2, nullptr, cW2T, cb2, S1, NC);
  embed_stage1<19><<<NV, 64, 0, stream>>>(var_x, vln_g, vln_b, vW1, vb1, S2, NV);
  wmma_gemm<64, 0, true, true><<<gemmGrid(NV), blk, 0, stream>>>(S2, nullptr, vW2T, vb2, S0, NV);

  // ---- v2c conv: left=var(S0), right=cons(S1), tgt=c_idx, src=v_idx ----
  wmma_gemm<64, 0, false, true><<<gemmGrid(NC), blk, 0, stream>>>(S1, nullptr, v2c_WlT, v2c.bl, S2, NC);   // li
  wmma_gemm<64, 0, false, false><<<gemmGrid(NV), blk, 0, stream>>>(S0, nullptr, v2c_WrT, nullptr, S3, NV); // rj
  fill0<<<dim3((unsigned)((NC * 64 + 255) / 256)), blk, 0, stream>>>(S4, NC * 64);
  edge_conv<<<edgeGrid, blk, 0, stream>>>(S2, S3, c_idx, v_idx, eln_b, v2c.We,
                                          v2c.fln_g, v2c.fln_b, v2c_WfT, v2c.bf, S4, E);
  ln64<<<NC, 64, 0, stream>>>(S4, v2c.pln_g, v2c.pln_b, S2, NC);
  wmma_gemm<64, 64, true, true><<<gemmGrid(NC), blk, 0, stream>>>(S2, S1, v2c_Wo1T, v2c.bo1, S4, NC);
  wmma_gemm<64, 0, false, true><<<gemmGrid(NC), blk, 0, stream>>>(S4, nullptr, v2c_Wo2T, v2c.bo2, S1, NC); // cons2

  // ---- c2v conv: left=cons2(S1), right=var(S0), tgt=v_idx, src=c_idx ----
  wmma_gemm<64, 0, false, true><<<gemmGrid(NV), blk, 0, stream>>>(S0, nullptr, c2v_WlT, c2v.bl, S3, NV);   // li
  wmma_gemm<64, 0, false, false><<<gemmGrid(NC), blk, 0, stream>>>(S1, nullptr, c2v_WrT, nullptr, S2, NC); // rj
  fill0<<<dim3((unsigned)((NV * 64 + 255) / 256)), blk, 0, stream>>>(S4, NV * 64);
  edge_conv<<<edgeGrid, blk, 0, stream>>>(S3, S2, v_idx, c_idx, eln_b, c2v.We,
                                          c2v.fln_g, c2v.fln_b, c2v_WfT, c2v.bf, S4, E);
  ln64<<<NV, 64, 0, stream>>>(S4, c2v.pln_g, c2v.pln_b, S3, NV);
  wmma_gemm<64, 64, true, true><<<gemmGrid(NV), blk, 0, stream>>>(S3, S0, c2v_Wo1T, c2v.bo1, S2, NV);
  wmma_gemm<64, 0, false, true><<<gemmGrid(NV), blk, 0, stream>>>(S2, nullptr, c2v_Wo2T, c2v.bo2, S0, NV); // var2

  // ---- Output head ----
  wmma_gemm<64, 0, true, true><<<gemmGrid(NV), blk, 0, stream>>>(S0, nullptr, Wout1T, bout1, S2, NV);
  out_dot<<<dim3((unsigned)((NV + 255) / 256)), blk, 0, stream>>>(S2, Wout2, (float*)d_out, NV);
}